// LSTMWithAttention_69157563400194
// MI455X (gfx1250) — compile-verified
//
#include <hip/hip_runtime.h>
#include <hip/hip_bf16.h>
#include <math.h>

#define HDIM 2048
#define VDIM 128000

typedef float v2f __attribute__((ext_vector_type(2)));
typedef float v8f __attribute__((ext_vector_type(8)));

__device__ __forceinline__ float sigf(float x) { return 1.0f / (1.0f + expf(-x)); }

// Accumulate a 16-row x K GEMV tile into a 16x16 f32 WMMA accumulator.
// Wrow = W + row*ldk + 4*hi  (row = rowbase + (lane&15), hi = lane>>4)
// xvec = x + 4*hi            (vector broadcast across all 16 N columns)
// Per 8 K columns: one dense b128 A load, one broadcast b128 B load, two WMMAs.
__device__ __forceinline__ v8f gemv_wmma_acc(const float* __restrict__ Wrow,
                                             const float* __restrict__ xvec,
                                             int K, v8f c)
{
    for (int k = 0; k < K; k += 8) {
        const float4 a = *reinterpret_cast<const float4*>(Wrow + k);
        const float4 b = *reinterpret_cast<const float4*>(xvec + k);
        v2f a0; a0[0] = a.x; a0[1] = a.y;
        v2f b0; b0[0] = b.x; b0[1] = b.y;
        c = __builtin_amdgcn_wmma_f32_16x16x4_f32(false, a0, false, b0,
                                                  (short)0, c, false, false);
        v2f a1; a1[0] = a.z; a1[1] = a.w;
        v2f b1; b1[0] = b.z; b1[1] = b.w;
        c = __builtin_amdgcn_wmma_f32_16x16x4_f32(false, a1, false, b1,
                                                  (short)0, c, false, false);
    }
    return c;
}

// Extract replicated GEMV result: all N columns equal; lanes 0..15 hold rows 0..7
// in c[0..7], lanes 16..31 hold rows 8..15.
__device__ __forceinline__ float comp8(v8f c, int j)
{
    switch (j) {
        case 0: return c[0]; case 1: return c[1]; case 2: return c[2];
        case 3: return c[3]; case 4: return c[4]; case 5: return c[5];
        case 6: return c[6]; default: return c[7];
    }
}

// gates = W_ih @ emb[tok] + W_hh @ h0 + b_ih + b_hh   (8192 rows, K=2048 twice)
__global__ __launch_bounds__(256) void k_gates(
    const int* __restrict__ tok, const float* __restrict__ h0,
    const float* __restrict__ emb,
    const float* __restrict__ W_ih, const float* __restrict__ W_hh,
    const float* __restrict__ b_ih, const float* __restrict__ b_hh,
    float* __restrict__ gates)
{
    const int lane    = threadIdx.x & 31;
    const int wave    = threadIdx.x >> 5;
    const int rowbase = blockIdx.x * 128 + wave * 16;
    const int row     = rowbase + (lane & 15);
    const int hi4     = (lane >> 4) * 4;

    const float* x = emb + (size_t)tok[0] * HDIM;

    v8f c = {0.f, 0.f, 0.f, 0.f, 0.f, 0.f, 0.f, 0.f};
    c = gemv_wmma_acc(W_ih + (size_t)row * HDIM + hi4, x  + hi4, HDIM, c);
    c = gemv_wmma_acc(W_hh + (size_t)row * HDIM + hi4, h0 + hi4, HDIM, c);

    const int   j   = lane & 7;
    const float val = comp8(c, j);
    if ((lane & 15) < 8) {
        const int r = rowbase + ((lane < 16) ? j : (8 + j));
        gates[r] = val + b_ih[r] + b_hh[r];
    }
}

// LSTM cell elementwise; emits h_new, c_new, attn_w (==1 exactly: softmax over L=1),
// and feat = concat(h1, h1) since context == h1.
__global__ __launch_bounds__(256) void k_lstm(
    const float* __restrict__ gates, const float* __restrict__ c0,
    float* __restrict__ out, float* __restrict__ feat)
{
    const int i = blockIdx.x * blockDim.x + threadIdx.x;
    if (i >= HDIM) return;
    const float ig = sigf(gates[i]);
    const float fg = sigf(gates[HDIM + i]);
    const float gg = tanhf(gates[2 * HDIM + i]);
    const float og = sigf(gates[3 * HDIM + i]);
    const float c1 = fg * c0[i] + ig * gg;
    const float h1 = og * tanhf(c1);
    out[VDIM + i]        = h1;   // h_new
    out[VDIM + HDIM + i] = c1;   // c_new
    feat[i]        = h1;
    feat[HDIM + i] = h1;
    if (i == 0) out[VDIM + 2 * HDIM] = 1.0f;  // attn_w: softmax over a single score
}

// logits = W_out @ feat + b_out   (128000 rows, K=4096) -> d_out[0:V]
__global__ __launch_bounds__(256) void k_logits(
    const float* __restrict__ W_out, const float* __restrict__ b_out,
    const float* __restrict__ feat, float* __restrict__ logits)
{
    const int lane    = threadIdx.x & 31;
    const int wave    = threadIdx.x >> 5;
    const int rowbase = blockIdx.x * 128 + wave * 16;
    const int row     = rowbase + (lane & 15);
    const int hi4     = (lane >> 4) * 4;

    v8f c = {0.f, 0.f, 0.f, 0.f, 0.f, 0.f, 0.f, 0.f};
    c = gemv_wmma_acc(W_out + (size_t)row * (2 * HDIM) + hi4, feat + hi4, 2 * HDIM, c);

    const int   j   = lane & 7;
    const float val = comp8(c, j);
    if ((lane & 15) < 8) {
        const int r = rowbase + ((lane < 16) ? j : (8 + j));
        logits[r] = val + b_out[r];
    }
}

// Per-block online-softmax partials over 1024 logits each (125 blocks x 1024 = 128000).
__global__ __launch_bounds__(1024) void k_partial(
    const float* __restrict__ logits, float* __restrict__ pm, float* __restrict__ ps)
{
    __shared__ float sm[1024];
    const int t = threadIdx.x;
    const float v = logits[blockIdx.x * 1024 + t];
    sm[t] = v; __syncthreads();
    for (int s = 512; s > 0; s >>= 1) { if (t < s) sm[t] = fmaxf(sm[t], sm[t + s]); __syncthreads(); }
    const float m = sm[0]; __syncthreads();
    sm[t] = expf(v - m); __syncthreads();
    for (int s = 512; s > 0; s >>= 1) { if (t < s) sm[t] += sm[t + s]; __syncthreads(); }
    if (t == 0) { pm[blockIdx.x] = m; ps[blockIdx.x] = sm[0]; }
}

// Combine 125 partials -> log-sum-exp scalar.
__global__ __launch_bounds__(128) void k_combine(
    const float* __restrict__ pm, const float* __restrict__ ps,
    float* __restrict__ lse, int n)
{
    __shared__ float sm[128], ss[128];
    const int t = threadIdx.x;
    const float m = (t < n) ? pm[t] : -INFINITY;
    sm[t] = m; __syncthreads();
    for (int s = 64; s > 0; s >>= 1) { if (t < s) sm[t] = fmaxf(sm[t], sm[t + s]); __syncthreads(); }
    const float M = sm[0];
    ss[t] = (t < n) ? ps[t] * expf(m - M) : 0.0f; __syncthreads();
    for (int s = 64; s > 0; s >>= 1) { if (t < s) ss[t] += ss[t + s]; __syncthreads(); }
    if (t == 0) lse[0] = M + logf(ss[0]);
}

// logp = logits - lse, in place in d_out.
__global__ __launch_bounds__(256) void k_logp(float* __restrict__ out,
                                              const float* __restrict__ lse)
{
    const int i = blockIdx.x * blockDim.x + threadIdx.x;
    if (i < VDIM) out[i] -= lse[0];
}

extern "C" void kernel_launch(void* const* d_in, const int* in_sizes, int n_in,
                              void* d_out, int out_size, void* d_ws, size_t ws_size,
                              hipStream_t stream)
{
    (void)in_sizes; (void)n_in; (void)out_size; (void)ws_size;
    const int*   tok   = (const int*)  d_in[0];
    const float* h0    = (const float*)d_in[1];
    const float* c0    = (const float*)d_in[2];
    const float* emb   = (const float*)d_in[3];
    const float* W_ih  = (const float*)d_in[4];
    const float* W_hh  = (const float*)d_in[5];
    const float* b_ih  = (const float*)d_in[6];
    const float* b_hh  = (const float*)d_in[7];
    // d_in[8..10] (W_attn, b_attn, v) eliminated analytically: softmax over L=1 == 1.
    const float* W_out = (const float*)d_in[11];
    const float* b_out = (const float*)d_in[12];

    float* out   = (float*)d_out;
    float* ws    = (float*)d_ws;
    float* gates = ws;          // 8192 floats
    float* feat  = ws + 8192;   // 4096 floats (16B-aligned offset)
    float* pm    = ws + 12288;  // 125 floats
    float* ps    = ws + 12416;  // 125 floats
    float* lse   = ws + 12544;  // 1 float

    k_gates  <<<  64, 256, 0, stream>>>(tok, h0, emb, W_ih, W_hh, b_ih, b_hh, gates);
    k_lstm   <<<   8, 256, 0, stream>>>(gates, c0, out, feat);
    k_logits <<<1000, 256, 0, stream>>>(W_out, b_out, feat, out);
    k_partial<<< 125, 1024, 0, stream>>>(out, pm, ps);
    k_combine<<<   1, 128, 0, stream>>>(pm, ps, lse, 125);
    k_logp   <<< 500, 256, 0, stream>>>(out, lse);
}